// GATLayer_82463372084005
// MI455X (gfx1250) — compile-verified
//
#include <hip/hip_runtime.h>
#include <hip/hip_bf16.h>
#include <math.h>

#define N_NODES 100000
#define N_EDGES 1600000
#define IN_DIM  128
#define OUT_DIM 64
#define LEAKY   0.01f

typedef float v2f __attribute__((ext_vector_type(2)));
typedef float v8f __attribute__((ext_vector_type(8)));

// ---------- order-preserving float <-> uint map for atomic max ----------
__device__ __forceinline__ unsigned fkey(float f) {
    unsigned u = __float_as_uint(f);
    return (u & 0x80000000u) ? ~u : (u | 0x80000000u);
}
__device__ __forceinline__ float keyf(unsigned k) {
    unsigned u = (k & 0x80000000u) ? (k & 0x7FFFFFFFu) : ~k;
    return __uint_as_float(u);
}
// key(-inf): bits(-inf)=0xFF800000 (sign set) -> ~u = 0x007FFFFF
#define NEG_INF_KEY 0x007FFFFFu

// ---------- 0) init: zero out, init mkey/denom ----------
__global__ void gat_init(float* __restrict__ out, unsigned* __restrict__ mkey,
                         float* __restrict__ denom) {
    int i = blockIdx.x * blockDim.x + threadIdx.x;
    if (i < N_NODES * OUT_DIM) out[i] = 0.0f;
    if (i < N_NODES) { mkey[i] = NEG_INF_KEY; denom[i] = 0.0f; }
}

// ---------- 1) z = h @ W^T via V_WMMA_F32_16X16X4_F32 ----------
// grid.x = N_NODES/16 row tiles (exact: 100000/16 = 6250)
// block = 128 threads = 4 waves; wave w computes output columns [16w, 16w+16)
__global__ void gat_gemm_wmma(const float* __restrict__ h,
                              const float* __restrict__ W,
                              float* __restrict__ z) {
    const int lane = threadIdx.x & 31;
    const int wave = threadIdx.x >> 5;
    const int row0 = blockIdx.x * 16;
    const int col0 = wave * 16;

    const int m     = lane & 15;        // A: row in tile; B: col in tile
    const int khalf = (lane >> 4) * 2;  // lanes 16-31 hold K+2, K+3

    const float* __restrict__ hrow = h + (size_t)(row0 + m) * IN_DIM; // A row
    const float* __restrict__ wrow = W + (size_t)(col0 + m) * IN_DIM; // B col = W row

    __builtin_prefetch(wrow, 0, 3);   // gfx1250 global_prefetch_b8

    v8f c = {};
#pragma unroll
    for (int k0 = 0; k0 < IN_DIM; k0 += 4) {
        v2f a, b;
        a.x = hrow[k0 + khalf];
        a.y = hrow[k0 + khalf + 1];
        b.x = wrow[k0 + khalf];
        b.y = wrow[k0 + khalf + 1];
        // 8 args: (neg_a, A, neg_b, B, c_mod, C, reuse_a, reuse_b)
        c = __builtin_amdgcn_wmma_f32_16x16x4_f32(
                false, a, false, b, (short)0, c, false, false);
    }

    // C/D layout: VGPR r holds M = r + 8*(lane>=16), N = lane&15
    const int n     = lane & 15;
    const int rbase = (lane >> 4) * 8;
#pragma unroll
    for (int r = 0; r < 8; ++r) {
        z[(size_t)(row0 + rbase + r) * OUT_DIM + col0 + n] = c[r];
    }
}

// ---------- 2) per-node projections s = z.a_src, t = z.a_dst ----------
__global__ void gat_proj(const float* __restrict__ z, const float* __restrict__ A,
                         float* __restrict__ s, float* __restrict__ t) {
    int i = blockIdx.x * blockDim.x + threadIdx.x;
    if (i >= N_NODES) return;
    const float4* zr = (const float4*)(z + (size_t)i * OUT_DIM);
    const float4* as = (const float4*)A;             // A[0,   0:64)
    const float4* ad = (const float4*)(A + OUT_DIM); // A[0, 64:128)
    float accs = 0.f, acct = 0.f;
#pragma unroll
    for (int k = 0; k < OUT_DIM / 4; ++k) {
        float4 v = zr[k], x = as[k], y = ad[k];
        accs += v.x * x.x + v.y * x.y + v.z * x.z + v.w * x.w;
        acct += v.x * y.x + v.y * y.y + v.z * y.z + v.w * y.w;
    }
    s[i] = accs;
    t[i] = acct;
}

// ---------- 3) e = leaky_relu(s[src]+t[dst]); segment max into mkey ----------
__global__ void gat_edge_logit(const int* __restrict__ src, const int* __restrict__ dst,
                               const float* __restrict__ s, const float* __restrict__ t,
                               float* __restrict__ eb, unsigned* __restrict__ mkey) {
    int e = blockIdx.x * blockDim.x + threadIdx.x;
    if (e >= N_EDGES) return;
    int si = src[e], di = dst[e];
    float v = s[si] + t[di];
    v = (v > 0.f) ? v : LEAKY * v;
    eb[e] = v;
    atomicMax(&mkey[di], fkey(v));
}

// ---------- 4) decode m, replace non-finite by 0 (in place over mkey) ----------
__global__ void gat_mfix(unsigned* __restrict__ mkey) {
    int i = blockIdx.x * blockDim.x + threadIdx.x;
    if (i >= N_NODES) return;
    float mv = keyf(mkey[i]);
    if (!isfinite(mv)) mv = 0.0f;
    ((float*)mkey)[i] = mv;
}

// ---------- 5) p = exp(e - m[dst]); denom[dst] += p ----------
__global__ void gat_p_denom(const int* __restrict__ dst, const float* __restrict__ m,
                            float* __restrict__ eb, float* __restrict__ denom) {
    int e = blockIdx.x * blockDim.x + threadIdx.x;
    if (e >= N_EDGES) return;
    int di = dst[e];
    float p = expf(eb[e] - m[di]);
    eb[e] = p;
    atomicAdd(&denom[di], p);
}

// ---------- 6) out[dst,:] += (p/denom[dst]) * z[src,:] ----------
// block = 256 threads = 8 waves; wave handles one edge; lane owns 2 columns
__global__ void gat_scatter(const int* __restrict__ src, const int* __restrict__ dst,
                            const float* __restrict__ p, const float* __restrict__ denom,
                            const float* __restrict__ z, float* __restrict__ out) {
    const int lane = threadIdx.x & 31;
    const int wave = threadIdx.x >> 5;
    const int e = blockIdx.x * 8 + wave;
    if (e >= N_EDGES) return;
    const int si = src[e], di = dst[e];
    const float alpha = p[e] / denom[di];
    float2 v = ((const float2*)(z + (size_t)si * OUT_DIM))[lane];
    float* o = out + (size_t)di * OUT_DIM + 2 * lane;
    atomicAdd(o + 0, alpha * v.x);
    atomicAdd(o + 1, alpha * v.y);
}

extern "C" void kernel_launch(void* const* d_in, const int* in_sizes, int n_in,
                              void* d_out, int out_size, void* d_ws, size_t ws_size,
                              hipStream_t stream) {
    const float* h  = (const float*)d_in[0];
    const int* src  = (const int*)d_in[1];
    const int* dst  = (const int*)d_in[2];
    const float* W  = (const float*)d_in[3];
    const float* A  = (const float*)d_in[4];
    float* out = (float*)d_out;

    char* ws = (char*)d_ws;
    float*    z     = (float*)   (ws);                 // 25,600,000 B
    float*    s     = (float*)   (ws + 25600000);      //    400,000 B
    float*    t     = (float*)   (ws + 26000000);      //    400,000 B
    unsigned* mkey  = (unsigned*)(ws + 26400000);      //    400,000 B (reused as float m)
    float*    denom = (float*)   (ws + 26800000);      //    400,000 B
    float*    eb    = (float*)   (ws + 27200000);      //  6,400,000 B (e then p)

    const int node_blocks  = (N_NODES + 255) / 256;          // 391
    const int edge_blocks  = (N_EDGES + 255) / 256;          // 6250
    const int init_blocks  = (N_NODES * OUT_DIM + 255) / 256;

    gat_init<<<init_blocks, 256, 0, stream>>>(out, mkey, denom);
    gat_gemm_wmma<<<N_NODES / 16, 128, 0, stream>>>(h, W, z);
    gat_proj<<<node_blocks, 256, 0, stream>>>(z, A, s, t);
    gat_edge_logit<<<edge_blocks, 256, 0, stream>>>(src, dst, s, t, eb, mkey);
    gat_mfix<<<node_blocks, 256, 0, stream>>>(mkey);
    gat_p_denom<<<edge_blocks, 256, 0, stream>>>(dst, (const float*)mkey, eb, denom);
    gat_scatter<<<(N_EDGES + 7) / 8, 256, 0, stream>>>(src, dst, eb, denom, z, out);
}